// dualGAT_53532472377640
// MI455X (gfx1250) — compile-verified
//
#include <hip/hip_runtime.h>
#include <hip/hip_bf16.h>

typedef __attribute__((ext_vector_type(2))) float v2f;
typedef __attribute__((ext_vector_type(8))) float v8f;

constexpr int Nn  = 20000;
constexpr int Ee  = 320000;
constexpr int ET  = Ee + Nn;   // edges + self loops
constexpr int KIN = 128;       // inner dim for every layer (F_IN = HEADS*HID = 128)

// ---------------------------------------------------------------------------
// CDNA5 device-scope float atomic max (global_atomic_max_num_f32, no return)
// ---------------------------------------------------------------------------
__device__ __forceinline__ void atomicMaxF32(float* addr, float val) {
    asm volatile("global_atomic_max_num_f32 %0, %1, off scope:SCOPE_DEV"
                 :: "v"(addr), "v"(val) : "memory");
}

// ---------------------------------------------------------------------------
// h = X[Nn x 128] @ W[128 x FOUT] with V_WMMA_F32_16X16X4_F32, fp32.
// One wave owns a 16-col panel and RB row tiles; K fully unrolled so every
// global load uses an immediate offset (no per-iteration address math).
// B fragment reused across RB row tiles.
// ---------------------------------------------------------------------------
template<int FOUT, int RB>
__global__ void gemm_wmma_f32(const float* __restrict__ X,
                              const float* __restrict__ W,
                              float* __restrict__ Hout) {
    const int lane = threadIdx.x & 31;
    const int wave = threadIdx.x >> 5;
    const int row0 = blockIdx.x * (16 * RB);
    const int col0 = wave * 16;
    const int m    = lane & 15;
    const int hi   = lane >> 4;        // 0: lanes 0-15, 1: lanes 16-31
    const int kh   = hi << 1;          // K sub-offset 0 or 2

    const float* Xp = X + (size_t)(row0 + m) * KIN + kh;   // + r*16*KIN + k (imm)
    const float* Wp = W + (size_t)kh * FOUT + col0 + m;    // + k*FOUT      (imm)

    v8f c[RB] = {};
#pragma unroll
    for (int k = 0; k < KIN; k += 4) {
        v2f b;
        b.x = Wp[(size_t)k * FOUT];
        b.y = Wp[(size_t)(k + 1) * FOUT];
#pragma unroll
        for (int r = 0; r < RB; ++r) {
            const float2 av = *(const float2*)(Xp + (size_t)r * 16 * KIN + k);
            v2f a; a.x = av.x; a.y = av.y;
            c[r] = __builtin_amdgcn_wmma_f32_16x16x4_f32(false, a, false, b,
                                                         (short)0, c[r], false, false);
        }
    }
    // C/D layout: VGPR i -> M = i + 8*hi
#pragma unroll
    for (int r = 0; r < RB; ++r)
#pragma unroll
        for (int i = 0; i < 8; ++i)
            Hout[(size_t)(row0 + r * 16 + 8 * hi + i) * FOUT + col0 + m] = c[r][i];
}

// ---------------------------------------------------------------------------
// Per-(node, head) attention scores: s = <h, a_s>, d = <h, a_d>
// ---------------------------------------------------------------------------
template<int HH, int C>
__global__ void node_scores(const float* __restrict__ Hm,
                            const float* __restrict__ a_s,
                            const float* __restrict__ a_d,
                            float* __restrict__ s, float* __restrict__ d) {
    int t = blockIdx.x * blockDim.x + threadIdx.x;
    if (t >= Nn * HH) return;
    int node = t / HH, hd = t - node * HH;
    const float* hp  = Hm + (size_t)(node * HH + hd) * C;
    const float* asp = a_s + hd * C;
    const float* adp = a_d + hd * C;
    float ss = 0.f, dd = 0.f;
#pragma unroll 8
    for (int c = 0; c < C; ++c) { float hv = hp[c]; ss += hv * asp[c]; dd += hv * adp[c]; }
    s[t] = ss; d[t] = dd;
}

// ---------------------------------------------------------------------------
// Init accumulators: out = 0, m = -inf, sum = 0
// ---------------------------------------------------------------------------
__global__ void init_accum(float* __restrict__ out, float* __restrict__ m,
                           float* __restrict__ sum, int nF, int nH) {
    int t = blockIdx.x * blockDim.x + threadIdx.x;
    if (t < nF) out[t] = 0.f;
    if (t < nH) { m[t] = -3.0e38f; sum[t] = 0.f; }
}

__device__ __forceinline__ void edge_nodes(int e, const int* __restrict__ src,
                                           const int* __restrict__ dst,
                                           int& u, int& v) {
    if (e < Ee) { u = src[e]; v = dst[e]; } else { u = v = e - Ee; }
}

// ---------------------------------------------------------------------------
// Pass 1: segment max of LeakyReLU(s[src]+d[dst]) over dst
// ---------------------------------------------------------------------------
template<int HH>
__global__ void edge_max_k(const int* __restrict__ src, const int* __restrict__ dst,
                           const float* __restrict__ s, const float* __restrict__ d,
                           float* __restrict__ m) {
    int t = blockIdx.x * blockDim.x + threadIdx.x;
    if (t >= ET * HH) return;
    int e = t / HH, hd = t - e * HH;   // HH constexpr -> shift/mask
    int u, v; edge_nodes(e, src, dst, u, v);
    float val = s[u * HH + hd] + d[v * HH + hd];
    val = val > 0.f ? val : 0.2f * val;
    atomicMaxF32(&m[v * HH + hd], val);
}

// ---------------------------------------------------------------------------
// Pass 2: ex = exp(e - m[dst]); sum[dst] += ex
// ---------------------------------------------------------------------------
template<int HH>
__global__ void edge_exp_k(const int* __restrict__ src, const int* __restrict__ dst,
                           const float* __restrict__ s, const float* __restrict__ d,
                           const float* __restrict__ m, float* __restrict__ sum,
                           float* __restrict__ ex) {
    int t = blockIdx.x * blockDim.x + threadIdx.x;
    if (t >= ET * HH) return;
    int e = t / HH, hd = t - e * HH;
    int u, v; edge_nodes(e, src, dst, u, v);
    float val = s[u * HH + hd] + d[v * HH + hd];
    val = val > 0.f ? val : 0.2f * val;
    float exv = __expf(val - m[v * HH + hd]);
    ex[t] = exv;
    atomicAdd(&sum[v * HH + hd], exv);
}

// ---------------------------------------------------------------------------
// Pass 3: out[dst] += h[src] * (ex / sum[dst]); one thread per (edge, head),
// float4 gathers of L2-resident h, C atomics amortize each index fetch.
// ---------------------------------------------------------------------------
template<int HH, int C>
__global__ void edge_scatter(const int* __restrict__ src, const int* __restrict__ dst,
                             const float* __restrict__ Hm, const float* __restrict__ ex,
                             const float* __restrict__ sum, float* __restrict__ out) {
    int t = blockIdx.x * blockDim.x + threadIdx.x;
    if (t >= ET * HH) return;
    int e = t / HH, hd = t - e * HH;
    int u, v; edge_nodes(e, src, dst, u, v);
    float alpha = ex[t] / sum[v * HH + hd];
    const float* hp = Hm + (size_t)(u * HH + hd) * C;
    float* op       = out + (size_t)(v * HH + hd) * C;
#pragma unroll
    for (int c = 0; c < C; c += 4) {
        float4 hv = *(const float4*)(hp + c);
        atomicAdd(op + c + 0, hv.x * alpha);
        atomicAdd(op + c + 1, hv.y * alpha);
        atomicAdd(op + c + 2, hv.z * alpha);
        atomicAdd(op + c + 3, hv.w * alpha);
    }
}

// ---------------------------------------------------------------------------
// Inter-layer: x_next = relu(out + b)
// ---------------------------------------------------------------------------
template<int F>
__global__ void bias_relu(const float* __restrict__ out, const float* __restrict__ b,
                          float* __restrict__ xn) {
    int t = blockIdx.x * blockDim.x + threadIdx.x;
    if (t >= Nn * F) return;
    float v = out[t] + b[t & (F - 1)];
    xn[t] = v > 0.f ? v : 0.f;
}

// ---------------------------------------------------------------------------
// Final: y (+)= 0.5 * sigmoid(out + b)
// ---------------------------------------------------------------------------
template<int F>
__global__ void bias_sigmoid_accum(const float* __restrict__ out,
                                   const float* __restrict__ b,
                                   float* __restrict__ y, int first) {
    int t = blockIdx.x * blockDim.x + threadIdx.x;
    if (t >= Nn * F) return;
    float v  = out[t] + b[t & (F - 1)];
    float sg = 1.f / (1.f + __expf(-v));
    float r  = 0.5f * sg;
    if (!first) r += y[t];
    y[t] = r;
}

static inline int gr(int n) { return (n + 255) / 256; }

extern "C" void kernel_launch(void* const* d_in, const int* in_sizes, int n_in,
                              void* d_out, int out_size, void* d_ws, size_t ws_size,
                              hipStream_t stream) {
    (void)in_sizes; (void)n_in; (void)out_size; (void)ws_size;

    float* ws   = (float*)d_ws;
    float* xbuf = ws;                            // Nn*128
    float* hbuf = xbuf + (size_t)Nn * 128;       // Nn*128
    float* obuf = hbuf + (size_t)Nn * 128;       // Nn*128
    float* sbuf = obuf + (size_t)Nn * 128;       // Nn*4
    float* dbuf = sbuf + (size_t)Nn * 4;         // Nn*4
    float* mbuf = dbuf + (size_t)Nn * 4;         // Nn*4
    float* sumb = mbuf + (size_t)Nn * 4;         // Nn*4
    float* exb  = sumb + (size_t)Nn * 4;         // ET*4

    constexpr int RB = 2;                        // row tiles per wave
    constexpr int GEMM_BLOCKS = Nn / (16 * RB);  // 20000/32 = 625

    for (int br = 0; br < 2; ++br) {
        const float* xin  = (const float*)d_in[br == 0 ? 0 : 2];
        const int*   ei   = (const int*)  d_in[br == 0 ? 1 : 3];
        const int*   srcA = ei;
        const int*   dstA = ei + Ee;
        const int    pbase = 4 + br * 16;

        for (int l = 0; l < 4; ++l) {
            const float* W   = (const float*)d_in[pbase + l * 4 + 0];
            const float* a_s = (const float*)d_in[pbase + l * 4 + 1];
            const float* a_d = (const float*)d_in[pbase + l * 4 + 2];
            const float* bb  = (const float*)d_in[pbase + l * 4 + 3];
            const float* xcur = (l == 0) ? xin : xbuf;
            const int Fout = (l == 3) ? 64 : 128;
            const int Hh   = (l == 3) ? 1  : 4;

            if (l < 3) {
                gemm_wmma_f32<128, RB><<<GEMM_BLOCKS, 8 * 32, 0, stream>>>(xcur, W, hbuf);
                node_scores<4, 32><<<gr(Nn * 4), 256, 0, stream>>>(hbuf, a_s, a_d, sbuf, dbuf);
            } else {
                gemm_wmma_f32<64, RB><<<GEMM_BLOCKS, 4 * 32, 0, stream>>>(xcur, W, hbuf);
                node_scores<1, 64><<<gr(Nn), 256, 0, stream>>>(hbuf, a_s, a_d, sbuf, dbuf);
            }

            init_accum<<<gr(Nn * Fout), 256, 0, stream>>>(obuf, mbuf, sumb, Nn * Fout, Nn * Hh);

            if (l < 3) {
                edge_max_k<4><<<gr(ET * 4), 256, 0, stream>>>(srcA, dstA, sbuf, dbuf, mbuf);
                edge_exp_k<4><<<gr(ET * 4), 256, 0, stream>>>(srcA, dstA, sbuf, dbuf, mbuf, sumb, exb);
                edge_scatter<4, 32><<<gr(ET * 4), 256, 0, stream>>>(srcA, dstA, hbuf, exb, sumb, obuf);
                bias_relu<128><<<gr(Nn * 128), 256, 0, stream>>>(obuf, bb, xbuf);
            } else {
                edge_max_k<1><<<gr(ET), 256, 0, stream>>>(srcA, dstA, sbuf, dbuf, mbuf);
                edge_exp_k<1><<<gr(ET), 256, 0, stream>>>(srcA, dstA, sbuf, dbuf, mbuf, sumb, exb);
                edge_scatter<1, 64><<<gr(ET), 256, 0, stream>>>(srcA, dstA, hbuf, exb, sumb, obuf);
                bias_sigmoid_accum<64><<<gr(Nn * 64), 256, 0, stream>>>(obuf, bb, (float*)d_out,
                                                                        br == 0 ? 1 : 0);
            }
        }
    }
}